// HH_46729244181187
// MI455X (gfx1250) — compile-verified
//
#include <hip/hip_runtime.h>
#include <hip/hip_bf16.h>
#include <stdint.h>

typedef __attribute__((ext_vector_type(2))) float        v2f;
typedef __attribute__((ext_vector_type(8))) float        v8f;
typedef __attribute__((ext_vector_type(4))) unsigned int v4u;
typedef __attribute__((ext_vector_type(8))) int          v8i;
typedef __attribute__((ext_vector_type(4))) int          v4i;

#define SDIM   256
#define NREFL  258
#define NCHUNK 16
#define MSZ    (SDIM * SDIM)   // 65536 floats per 256x256 matrix
#define QROW   257             // LDS row stride (floats) -> bank-conflict free

// ---------------------------------------------------------------------------
// fp32 WMMA 16x16x4 wrapper (D = A x B + C), wave32.
// A: 16x4 (v0: K=0|K=2, v1: K=1|K=3 across lane halves), B: 4x16 mirrored.
// ---------------------------------------------------------------------------
__device__ __forceinline__ v8f wmma4(v2f a, v2f b, v8f c) {
  return __builtin_amdgcn_wmma_f32_16x16x4_f32(false, a, false, b, (short)0, c,
                                               false, false);
}

// ---------------------------------------------------------------------------
// Kernel 1: each block builds P_c = prod_{k in chunk c} H_k  (in order),
// Q kept in LDS (row-per-thread; stride 257 floats avoids bank conflicts).
// Q <- Q @ H_k  ==  Q - (2/(v.v+eps)) * (Q v) v^T   (row-local update).
// ---------------------------------------------------------------------------
__global__ void __launch_bounds__(256, 1)
hh_partials(const float* __restrict__ vecs, float* __restrict__ parts) {
  extern __shared__ float smem[];
  float* Qs = smem;                 // QROW * 256 floats
  float* u  = smem + QROW * SDIM;   // 256 floats
  const int tid = threadIdx.x;      // 0..255 (row owner)
  const int c   = blockIdx.x;       // chunk id
  const int s   = (c * NREFL) / NCHUNK;
  const int e   = ((c + 1) * NREFL) / NCHUNK;

  float* row = Qs + tid * QROW;
  for (int j = 0; j < SDIM; ++j) row[j] = (tid == j) ? 1.0f : 0.0f;
  __syncthreads();

  for (int k = s; k < e; ++k) {
    u[tid] = vecs[k * SDIM + tid];
    __syncthreads();
    float inner = 0.0f;
    for (int j = 0; j < SDIM; ++j) inner = fmaf(u[j], u[j], inner);
    const float scale = 2.0f / (inner + 1e-16f);
    float w = 0.0f;
    for (int j = 0; j < SDIM; ++j) w = fmaf(row[j], u[j], w);
    const float sw = scale * w;
    for (int j = 0; j < SDIM; ++j) row[j] = fmaf(-sw, u[j], row[j]);
    __syncthreads();  // protect u[] before next iteration overwrites it
  }

  // coalesced writeout (consecutive tids -> consecutive addresses)
  float* dst = parts + (size_t)c * MSZ;
  for (int idx = tid; idx < MSZ; idx += 256)
    dst[idx] = Qs[(idx >> 8) * QROW + (idx & 255)];
}

// ---------------------------------------------------------------------------
// Kernel 2: pairwise combine  R_p = P_{2p} @ P_{2p+1}   (256x256x256 fp32,
// WMMA). 4 blocks x 8 waves per pair; each wave owns a 16x128 tile
// (8 x v8f = 64 accumulator VGPRs -> no spills).
// transpose!=0: final level, store Q into k-paired transposed layout Qp:
//   Qp[(n>>1)*512 + m*2 + (n&1)] = Q[m][n]   (== Q^T with K pairs adjacent)
// ---------------------------------------------------------------------------
__global__ void __launch_bounds__(256, 1)
hh_combine(const float* __restrict__ src, float* __restrict__ dst,
           int transpose) {
  const int tid  = threadIdx.x;
  const int lane = tid & 31;
  const int l15  = lane & 15;
  const int h    = lane >> 4;                             // lane half
  const int pair = blockIdx.x >> 2;
  const int task = ((blockIdx.x & 3) << 3) + (tid >> 5);  // 0..31
  const int m0   = (task >> 1) << 4;                      // row strip
  const int n0   = (task & 1) << 7;                       // col half

  const float* A = src + (size_t)(2 * pair) * MSZ;
  const float* B = src + (size_t)(2 * pair + 1) * MSZ;

  const v8f zero = {0.f, 0.f, 0.f, 0.f, 0.f, 0.f, 0.f, 0.f};
  v8f acc[8];
#pragma unroll
  for (int t = 0; t < 8; ++t) acc[t] = zero;

  const float* ap = A + (size_t)(m0 + l15) * SDIM + 2 * h;
#pragma unroll 2
  for (int kb = 0; kb < 64; ++kb) {
    const int k = kb * 4;
    const v2f a = *(const v2f*)(ap + k);
    const int kp = k + 2 * h;
    const float* b0p = B + (size_t)kp * SDIM + n0 + l15;
#pragma unroll
    for (int t = 0; t < 8; ++t) {
      v2f b;
      b.x = b0p[t * 16];
      b.y = b0p[SDIM + t * 16];
      acc[t] = wmma4(a, b, acc[t]);
    }
  }

  if (!transpose) {
    float* d = dst + (size_t)pair * MSZ + (size_t)(m0 + 8 * h) * SDIM + n0 + l15;
#pragma unroll
    for (int t = 0; t < 8; ++t) {
#pragma unroll
      for (int v = 0; v < 8; ++v) {
        d[(size_t)v * SDIM + t * 16] = acc[t][v];
      }
    }
  } else {
#pragma unroll
    for (int t = 0; t < 8; ++t) {
      const int n = n0 + t * 16 + l15;
      const int base = (n >> 1) * 512 + (n & 1);
#pragma unroll
      for (int v = 0; v < 8; ++v) {
        const int m = m0 + v + 8 * h;
        dst[base + m * 2] = acc[t][v];
      }
    }
  }
}

// ---------------------------------------------------------------------------
// Kernel 3: out = x @ Q^T.  Qp (256KB) staged whole into LDS via one TDM
// tensor_load_to_lds; 16 waves/block, each wave -> 16 rows x 128 cols.
// ---------------------------------------------------------------------------
__device__ __forceinline__ void tdm_load_qp(const float* gsrc,
                                            unsigned lds_off) {
  const unsigned long long ga = (unsigned long long)(uintptr_t)gsrc;
  v4u g0;
  g0.x = 1u;                                        // count=1 (valid D#)
  g0.y = lds_off;                                   // lds_addr (bytes)
  g0.z = (unsigned)(ga & 0xFFFFFFFFu);              // global_addr[31:0]
  g0.w = (unsigned)((ga >> 32) & 0x1FFFFFFu) | (2u << 30);  // [56:32] | type=2
  v8i g1;
  g1[0] = (int)(2u << 16);        // data_size = 2 (4 bytes); wg_mask = 0
  g1[1] = (int)(512u << 16);      // tensor_dim0 = 512 elements
  g1[2] = (int)(128u << 16);      // tensor_dim1 = 128
  g1[3] = (int)(512u << 16);      // tile_dim0 = 512
  g1[4] = (int)128;               // tile_dim1 = 128 (tile_dim2 = 0)
  g1[5] = (int)512;               // tensor_dim0_stride = 512
  g1[6] = 0;
  g1[7] = 0;
  const v4i gz4 = {0, 0, 0, 0};
  const v8i gz8 = {0, 0, 0, 0, 0, 0, 0, 0};
  __builtin_amdgcn_tensor_load_to_lds(g0, g1, gz4, gz4, gz8, 0);
}

__global__ void __launch_bounds__(512, 1)
hh_apply(const float* __restrict__ x, const float* __restrict__ Qp,
         float* __restrict__ out) {
  extern __shared__ float lqp[];  // 65536 floats = 256 KB (fits 320KB WGP LDS)
  const int tid = threadIdx.x;    // 0..511 (16 wave32s)

  if (tid < 32) {                 // wave 0 issues the DMA, waits, then barrier
    tdm_load_qp(Qp, (unsigned)(uintptr_t)lqp);
    __builtin_amdgcn_s_wait_tensorcnt(0);
  }
  __syncthreads();

  const int lane = tid & 31;
  const int l15  = lane & 15;
  const int h    = lane >> 4;
  const int wv   = tid >> 5;                           // 0..15
  const int r0   = blockIdx.x * 128 + (wv >> 1) * 16;  // 16-row strip
  const int n0   = (wv & 1) << 7;                      // col half (0 or 128)

  const v8f zero = {0.f, 0.f, 0.f, 0.f, 0.f, 0.f, 0.f, 0.f};
  v8f acc[8];
#pragma unroll
  for (int t = 0; t < 8; ++t) acc[t] = zero;

  const float* xp = x + (size_t)(r0 + l15) * SDIM + 2 * h;
  const float* lb = lqp + n0 * 2 + l15 * 2;

#pragma unroll 2
  for (int kb = 0; kb < 64; ++kb) {
    const v2f a = *(const v2f*)(xp + 4 * kb);         // A frag: global b64
    const float* lrow = lb + (2 * kb + h) * 512;      // B frags: LDS b64
#pragma unroll
    for (int t = 0; t < 8; ++t) {
      const v2f b = *(const v2f*)(lrow + 32 * t);
      acc[t] = wmma4(a, b, acc[t]);
    }
  }

  float* op = out + (size_t)(r0 + 8 * h) * SDIM + n0 + l15;
#pragma unroll
  for (int t = 0; t < 8; ++t) {
#pragma unroll
    for (int v = 0; v < 8; ++v) {
      __builtin_nontemporal_store(acc[t][v], &op[v * SDIM + t * 16]);
    }
  }
}

// ---------------------------------------------------------------------------
// Launch. ws layout (units of 256KB matrices): [0..15] partials, [16..23]
// level-1, [0..3] level-2, [4..5] level-3, [6] = Qp. Needs ~6 MB scratch.
// ---------------------------------------------------------------------------
extern "C" void kernel_launch(void* const* d_in, const int* in_sizes, int n_in,
                              void* d_out, int out_size, void* d_ws,
                              size_t ws_size, hipStream_t stream) {
  const float* x    = (const float*)d_in[0];
  const float* vecs = (const float*)d_in[1];
  float* out = (float*)d_out;
  float* wsf = (float*)d_ws;

  float* mats = wsf;                       // 16 partial products
  float* lvlB = wsf + (size_t)16 * MSZ;    // 8 matrices
  float* lvlD = wsf + (size_t)4 * MSZ;     // 2 matrices (inside mats region)
  float* qp   = wsf + (size_t)6 * MSZ;     // final Q^T (k-paired layout)

  hh_partials<<<NCHUNK, 256, (QROW * SDIM + SDIM) * sizeof(float), stream>>>(
      vecs, mats);
  hh_combine<<<32, 256, 0, stream>>>(mats, lvlB, 0);  // 16 -> 8
  hh_combine<<<16, 256, 0, stream>>>(lvlB, mats, 0);  // 8  -> 4
  hh_combine<<<8, 256, 0, stream>>>(mats, lvlD, 0);   // 4  -> 2
  hh_combine<<<4, 256, 0, stream>>>(lvlD, qp, 1);     // 2  -> Qp (transposed)

  hh_apply<<<512, 512, MSZ * sizeof(float), stream>>>(x, qp, out);
}